// FairnessLossVariant_30219389895059
// MI455X (gfx1250) — compile-verified
//
#include <hip/hip_runtime.h>

// N = 384 matrices, fp32. All heavy O(N^3) work (the "envy matvecs" of the
// blocking-pair loss) runs through v_wmma_f32_16x16x4_f32 on the CDNA5 matrix
// pipe; VALU only builds the relu(x_j - x_k) A-tiles.

#define NN 384
#define NT 256   // 8 wave32 waves per block

typedef __attribute__((ext_vector_type(2))) float v2f;
typedef __attribute__((ext_vector_type(8))) float v8f;

// ---------------- block reductions (deterministic trees) ----------------
__device__ __forceinline__ float blockReduceSum(float v, float* red) {
    const int tid = threadIdx.x;
    red[tid] = v;
    __syncthreads();
    for (int s = NT / 2; s > 0; s >>= 1) {
        if (tid < s) red[tid] += red[tid + s];
        __syncthreads();
    }
    float r = red[0];
    __syncthreads();
    return r;
}

__device__ __forceinline__ float blockReduceMax(float v, float* red) {
    const int tid = threadIdx.x;
    red[tid] = v;
    __syncthreads();
    for (int s = NT / 2; s > 0; s >>= 1) {
        if (tid < s) red[tid] = fmaxf(red[tid], red[tid + s]);
        __syncthreads();
    }
    float r = red[0];
    __syncthreads();
    return r;
}

// ---------------- softmax over rows of m_hat + fused stats ----------------
// per row i: m_A[i,:], rnormA[i]=||m_A[i,:]||, fairA[i]=sum_j m_A[i,j]*(satA[i,j]-satB[j,i])
__global__ void row_softmax_kernel(const float* __restrict__ mhat,
                                   const float* __restrict__ satA,
                                   const float* __restrict__ satB,
                                   float* __restrict__ mA,
                                   float* __restrict__ rnormA,
                                   float* __restrict__ fairA) {
    __shared__ float xs[NN];
    __shared__ float red[NT];
    const int i = blockIdx.x, tid = threadIdx.x;
    const float* row = mhat + (size_t)i * NN;

    float lmax = -3.402823466e+38f;
    for (int j = tid; j < NN; j += NT) { float v = row[j]; xs[j] = v; lmax = fmaxf(lmax, v); }
    __syncthreads();
    const float mx = blockReduceMax(lmax, red);

    float lsum = 0.f;
    for (int j = tid; j < NN; j += NT) { float e = expf(xs[j] - mx); xs[j] = e; lsum += e; }
    __syncthreads();
    const float inv = 1.0f / blockReduceSum(lsum, red);

    float lsq = 0.f, lfair = 0.f;
    for (int j = tid; j < NN; j += NT) {
        float p = xs[j] * inv;
        mA[(size_t)i * NN + j] = p;
        lsq += p * p;
        lfair += p * (satA[(size_t)i * NN + j] - satB[(size_t)j * NN + i]);
    }
    const float ssq = blockReduceSum(lsq, red);
    const float sf  = blockReduceSum(lfair, red);
    if (tid == 0) { rnormA[i] = sqrtf(ssq); fairA[i] = sf; }
}

// ---------------- softmax over cols of m_hat + fused stats ----------------
// per col j: m_B[:,j] (written row-major and transposed), cnormB[j],
// fairB[j] = sum_k m_B[k,j]*(satB[j,k]-satA[k,j])
__global__ void col_softmax_kernel(const float* __restrict__ mhat,
                                   const float* __restrict__ satA,
                                   const float* __restrict__ satB,
                                   float* __restrict__ mB,    // row-major [k,j]
                                   float* __restrict__ mBt,   // mBt[j,k] = mB[k,j]
                                   float* __restrict__ cnormB,
                                   float* __restrict__ fairB) {
    __shared__ float xs[NN];
    __shared__ float red[NT];
    const int j = blockIdx.x, tid = threadIdx.x;

    float lmax = -3.402823466e+38f;
    for (int k = tid; k < NN; k += NT) { float v = mhat[(size_t)k * NN + j]; xs[k] = v; lmax = fmaxf(lmax, v); }
    __syncthreads();
    const float mx = blockReduceMax(lmax, red);

    float lsum = 0.f;
    for (int k = tid; k < NN; k += NT) { float e = expf(xs[k] - mx); xs[k] = e; lsum += e; }
    __syncthreads();
    const float inv = 1.0f / blockReduceSum(lsum, red);

    float lsq = 0.f, lfair = 0.f;
    for (int k = tid; k < NN; k += NT) {
        float p = xs[k] * inv;
        mB[(size_t)k * NN + j] = p;
        mBt[(size_t)j * NN + k] = p;
        lsq += p * p;
        lfair += p * (satB[(size_t)j * NN + k] - satA[(size_t)k * NN + j]);
    }
    const float ssq = blockReduceSum(lsq, red);
    const float sf  = blockReduceSum(lfair, red);
    if (tid == 0) { cnormB[j] = sqrtf(ssq); fairB[j] = sf; }
}

// ---------------- cosine dot: dotAB[i] = sum_j m_A[i,j]*m_B[j,i] ----------------
__global__ void cosdot_kernel(const float* __restrict__ mA,
                              const float* __restrict__ mBt,
                              float* __restrict__ dotAB) {
    __shared__ float red[NT];
    const int i = blockIdx.x, tid = threadIdx.x;
    float loc = 0.f;
    for (int j = tid; j < NN; j += NT)
        loc += mA[(size_t)i * NN + j] * mBt[(size_t)i * NN + j];
    const float s = blockReduceSum(loc, red);
    if (tid == 0) dotAB[i] = s;
}

// ---------------- envy matvec core: y[j] = sum_k relu(x[j]-x[k]) * w[k] ----------------
// One wave computes a 16-wide j-tile; inner K loop steps 4 via v_wmma_f32_16x16x4_f32.
// A-tile layout (ISA 16x4 f32): lane l holds M = l&15, K = (l>>4)*2 + vgpr.
// B is the broadcast of w[k0..k0+3] across all 16 N columns, so every D column
// equals the matvec result; lanes with (l&15)==0 extract it (M = vgpr + (l>>4)*8).
__device__ __forceinline__ void envy_compute(const float* __restrict__ xs,
                                             const float* __restrict__ ws,
                                             float* __restrict__ ys) {
    const int lane = threadIdx.x & 31;
    const int wave = threadIdx.x >> 5;
    const int m    = lane & 15;
    const int koff = (lane >> 4) * 2;
    for (int t = wave; t < NN / 16; t += NT / 32) {
        const int j0 = t * 16;
        const float xj = xs[j0 + m];
        v8f acc = {};
        for (int k0 = 0; k0 < NN; k0 += 4) {
            v2f a, b;
            a.x = fmaxf(xj - xs[k0 + koff], 0.0f);
            a.y = fmaxf(xj - xs[k0 + koff + 1], 0.0f);
            b.x = ws[k0 + koff];
            b.y = ws[k0 + koff + 1];
            acc = __builtin_amdgcn_wmma_f32_16x16x4_f32(
                false, a, false, b, (short)0, acc, false, false);
        }
        if (m == 0) {
            const int base = j0 + (lane >> 4) * 8;
#pragma unroll
            for (int v = 0; v < 8; ++v) ys[base + v] = acc[v];
        }
    }
}

// g(row r) stored: Gout[r,:] = envy(x = X[r,:], w = W[r*rs + k*es])
__global__ void envy_store_kernel(const float* __restrict__ X,
                                  const float* __restrict__ W,
                                  int rs, int es,
                                  float* __restrict__ Gout) {
    __shared__ float xs[NN], ws[NN], ys[NN];
    const int r = blockIdx.x, tid = threadIdx.x;
    for (int k = tid; k < NN; k += NT) {
        xs[k] = X[(size_t)r * NN + k];
        ws[k] = W[(size_t)r * rs + (size_t)k * es];
    }
    __syncthreads();
    envy_compute(xs, ws, ys);
    __syncthreads();
    for (int j = tid; j < NN; j += NT) Gout[(size_t)r * NN + j] = ys[j];
}

// y = envy(x = X[j,:], w = W[j*rs + k*es]); partial[j] = sum_i y[i] * G[i,j]
__global__ void envy_dot_kernel(const float* __restrict__ X,
                                const float* __restrict__ W,
                                int rs, int es,
                                const float* __restrict__ G,
                                float* __restrict__ partial) {
    __shared__ float xs[NN], ws[NN], ys[NN];
    __shared__ float red[NT];
    const int j = blockIdx.x, tid = threadIdx.x;
    for (int k = tid; k < NN; k += NT) {
        xs[k] = X[(size_t)j * NN + k];
        ws[k] = W[(size_t)j * rs + (size_t)k * es];
    }
    __syncthreads();
    envy_compute(xs, ws, ys);
    __syncthreads();
    float loc = 0.f;
    for (int i = tid; i < NN; i += NT) loc += ys[i] * G[(size_t)i * NN + j];
    const float s = blockReduceSum(loc, red);
    if (tid == 0) partial[j] = s;
}

// ---------------- final combine ----------------
__global__ void final_kernel(const float* __restrict__ rnormA,
                             const float* __restrict__ cnormB,
                             const float* __restrict__ dotAB,
                             const float* __restrict__ fairA,
                             const float* __restrict__ fairB,
                             const float* __restrict__ lsA,
                             const float* __restrict__ lsB,
                             float* __restrict__ out) {
    __shared__ float red[NT];
    const int tid = threadIdx.x;
    float c = 0.f, fa = 0.f, fb = 0.f, la = 0.f, lb = 0.f;
    for (int i = tid; i < NN; i += NT) {
        float ra = fmaxf(rnormA[i], 1e-12f);
        float cb = fmaxf(cnormB[i], 1e-12f);
        c  += dotAB[i] / (ra * cb);
        fa += fairA[i];
        fb += fairB[i];
        la += lsA[i];
        lb += lsB[i];
    }
    c  = blockReduceSum(c,  red);
    fa = blockReduceSum(fa, red);
    fb = blockReduceSum(fb, red);
    la = blockReduceSum(la, red);
    lb = blockReduceSum(lb, red);
    if (tid == 0) {
        const float invN2 = 1.0f / ((float)NN * (float)NN);
        float L_m  = 1.0f - c / (float)NN;          // c_AB == c_BA
        float LsA  = la * invN2;
        float LsB  = lb * invN2;
        float LfA  = fa * fa * invN2;
        float LfB  = fb * fb * invN2;
        out[0] = 1.0f * L_m + 0.5f * (0.7f * (LsA + LsB) + 0.01f * (LfA + LfB));
    }
}

extern "C" void kernel_launch(void* const* d_in, const int* in_sizes, int n_in,
                              void* d_out, int out_size, void* d_ws, size_t ws_size,
                              hipStream_t stream) {
    (void)in_sizes; (void)n_in; (void)out_size; (void)ws_size;
    const float* m_hat = (const float*)d_in[0];
    const float* sat_A = (const float*)d_in[1];
    const float* sat_B = (const float*)d_in[2];
    float* out = (float*)d_out;

    float* ws  = (float*)d_ws;
    float* mA  = ws;                 // N*N
    float* mB  = ws + 1 * NN * NN;   // N*N (row-major [k,j])
    float* mBt = ws + 2 * NN * NN;   // N*N (mBt[j,k] = mB[k,j])
    float* G   = ws + 3 * NN * NN;   // N*N (g_A, then reused for g_A')
    float* sm  = ws + 4 * NN * NN;   // small arrays
    float* rnormA = sm + 0 * NN;
    float* cnormB = sm + 1 * NN;
    float* dotAB  = sm + 2 * NN;
    float* fairA  = sm + 3 * NN;
    float* fairB  = sm + 4 * NN;
    float* lsA    = sm + 5 * NN;
    float* lsB    = sm + 6 * NN;

    row_softmax_kernel<<<NN, NT, 0, stream>>>(m_hat, sat_A, sat_B, mA, rnormA, fairA);
    col_softmax_kernel<<<NN, NT, 0, stream>>>(m_hat, sat_A, sat_B, mB, mBt, cnormB, fairB);
    cosdot_kernel     <<<NN, NT, 0, stream>>>(mA, mBt, dotAB);

    // L_s_A: g_A[i,:] = envy(satA row i, w = mA row i)
    envy_store_kernel <<<NN, NT, 0, stream>>>(sat_A, mA, NN, 1, G);
    // g_B[:,j] = envy(satB row j, w = mA column j); dot against g_A column j
    envy_dot_kernel   <<<NN, NT, 0, stream>>>(sat_B, mA, 1, NN, G, lsA);

    // L_s_B: g_A'[i,:] = envy(satB row i, w = mB column i = mBt row i)
    envy_store_kernel <<<NN, NT, 0, stream>>>(sat_B, mBt, NN, 1, G);
    // g_B'[:,j] = envy(satA row j, w = mB row j); dot against g_A' column j
    envy_dot_kernel   <<<NN, NT, 0, stream>>>(sat_A, mB, NN, 1, G, lsB);

    final_kernel<<<1, NT, 0, stream>>>(rnormA, cnormB, dotAB, fairA, fairB, lsA, lsB, out);
}